// TopKRouter_86406152061621
// MI455X (gfx1250) — compile-verified
//
#include <hip/hip_runtime.h>
#include <hip/hip_bf16.h>
#include <math.h>

#define HIDDEN_DIM 4096
#define NUM_EXPERTS 64
#define KCHUNK 32
#define TOK_PER_WG 128
#define THREADS 256
#define LPAD (NUM_EXPERTS + 2)   // padded logits row stride (floats)

#define RX_N ((TOK_PER_WG * KCHUNK) / THREADS)   // 16 floats/thread
#define RW_N ((NUM_EXPERTS * KCHUNK) / THREADS)  //  8 floats/thread

typedef __attribute__((ext_vector_type(2))) float v2f;
typedef __attribute__((ext_vector_type(8))) float v8f;

// ---------------------------------------------------------------------------
// Init: zero the global accumulator slots (counts[64], avg[64], entropy, gini)
// ---------------------------------------------------------------------------
__global__ void router_init(float* __restrict__ out, int nTok) {
    int i = blockIdx.x * blockDim.x + threadIdx.x;
    long base = 4l * nTok;
    if (i < 130) out[base + i] = 0.0f;
}

// ---------------------------------------------------------------------------
// Main: GEMM (f32 WMMA 16x16x4) with register-prefetch double buffering,
// then softmax + top-2 + per-block stat reduction
// ---------------------------------------------------------------------------
__global__ __launch_bounds__(THREADS) void router_main(
    const float* __restrict__ x, const float* __restrict__ W,
    float* __restrict__ out, int nTok)
{
    __shared__ float sX[TOK_PER_WG][KCHUNK];      // 16 KB token tile
    __shared__ float sW[NUM_EXPERTS][KCHUNK];     // 8 KB expert tile
    __shared__ float sL[TOK_PER_WG][LPAD];        // ~33 KB logits -> probs
    __shared__ float sCnt[NUM_EXPERTS];
    __shared__ float sEnt;

    const int tid  = threadIdx.x;
    const int wave = tid >> 5;
    const int lane = tid & 31;
    const long tokBase = (long)blockIdx.x * TOK_PER_WG;

    v8f acc0 = {0,0,0,0,0,0,0,0};
    v8f acc1 = {0,0,0,0,0,0,0,0};
    v8f acc2 = {0,0,0,0,0,0,0,0};
    v8f acc3 = {0,0,0,0,0,0,0,0};

    // A 16x4 f32 layout: lanes 0-15 -> M=lane, K={0,1}; lanes 16-31 -> M=lane-16, K={2,3}
    const int m  = lane & 15;          // M row (A) / N col (B)
    const int kb = (lane >> 4) << 1;   // K sub-offset {0, 2}

    // register staging for global->LDS (prefetch pipeline)
    float rx[RX_N];
    float rw[RW_N];

    auto load_chunk = [&](int kc) {
        #pragma unroll
        for (int i = 0; i < RX_N; ++i) {
            int f = tid + i * THREADS;
            rx[i] = x[(tokBase + f / KCHUNK) * HIDDEN_DIM + kc + (f % KCHUNK)];
        }
        #pragma unroll
        for (int i = 0; i < RW_N; ++i) {
            int f = tid + i * THREADS;
            rw[i] = W[(long)(f / KCHUNK) * HIDDEN_DIM + kc + (f % KCHUNK)];
        }
    };
    auto store_chunk = [&]() {
        #pragma unroll
        for (int i = 0; i < RX_N; ++i) {
            int f = tid + i * THREADS;
            sX[f / KCHUNK][f % KCHUNK] = rx[i];
        }
        #pragma unroll
        for (int i = 0; i < RW_N; ++i) {
            int f = tid + i * THREADS;
            sW[f / KCHUNK][f % KCHUNK] = rw[i];
        }
    };

    load_chunk(0);                       // prologue fetch

    for (int kc = 0; kc < HIDDEN_DIM; kc += KCHUNK) {
        store_chunk();                   // regs -> LDS (prev reads fenced by loop-end barrier)
        __syncthreads();

        if (kc + KCHUNK < HIDDEN_DIM)    // issue next chunk's HBM loads NOW;
            load_chunk(kc + KCHUNK);     // they complete while the WMMAs below run

        #pragma unroll
        for (int ks = 0; ks < KCHUNK; ks += 4) {
            v2f a  = *(const v2f*)&sX[wave * 16 + m][ks + kb];
            v2f b0 = *(const v2f*)&sW[ 0 + m][ks + kb];
            v2f b1 = *(const v2f*)&sW[16 + m][ks + kb];
            v2f b2 = *(const v2f*)&sW[32 + m][ks + kb];
            v2f b3 = *(const v2f*)&sW[48 + m][ks + kb];
            acc0 = __builtin_amdgcn_wmma_f32_16x16x4_f32(false, a, false, b0, (short)0, acc0, false, false);
            acc1 = __builtin_amdgcn_wmma_f32_16x16x4_f32(false, a, false, b1, (short)0, acc1, false, false);
            acc2 = __builtin_amdgcn_wmma_f32_16x16x4_f32(false, a, false, b2, (short)0, acc2, false, false);
            acc3 = __builtin_amdgcn_wmma_f32_16x16x4_f32(false, a, false, b3, (short)0, acc3, false, false);
        }
        __syncthreads();
    }

    // Scatter C tiles to LDS.  C 16x16 f32 layout: VGPR r, lanes 0-15 -> (M=r, N=lane),
    // lanes 16-31 -> (M=r+8, N=lane-16).
    {
        const int mbase = (lane >> 4) << 3;    // 0 or 8
        #pragma unroll
        for (int r = 0; r < 8; ++r) {
            int tr = wave * 16 + mbase + r;
            sL[tr][ 0 + m] = acc0[r];
            sL[tr][16 + m] = acc1[r];
            sL[tr][32 + m] = acc2[r];
            sL[tr][48 + m] = acc3[r];
        }
    }
    if (tid < NUM_EXPERTS) sCnt[tid] = 0.0f;
    if (tid == 0) sEnt = 0.0f;
    __syncthreads();

    // one thread per token: softmax, entropy, top-2
    if (tid < TOK_PER_WG) {
        float* row = sL[tid];
        float mx = -3.402823e38f;
        #pragma unroll
        for (int e = 0; e < NUM_EXPERTS; ++e) mx = fmaxf(mx, row[e]);
        float sum = 0.0f;
        #pragma unroll
        for (int e = 0; e < NUM_EXPERTS; ++e) sum += __expf(row[e] - mx);
        float inv = 1.0f / sum;

        float p1 = -1.0f, p2 = -1.0f, h = 0.0f;
        int i1 = 0, i2 = 0;
        #pragma unroll
        for (int e = 0; e < NUM_EXPERTS; ++e) {
            float p = __expf(row[e] - mx) * inv;
            row[e] = p;                         // probs in place for column sums
            h -= p * __logf(p + 1e-10f);
            if (p > p1)      { p2 = p1; i2 = i1; p1 = p; i1 = e; }
            else if (p > p2) { p2 = p;  i2 = e; }
        }

        long g = tokBase + tid;
        out[2 * g]     = (float)i1;
        out[2 * g + 1] = (float)i2;
        float wnorm = 1.0f / (p1 + p2);
        out[2l * nTok + 2 * g]     = p1 * wnorm;
        out[2l * nTok + 2 * g + 1] = p2 * wnorm;

        atomicAdd(&sCnt[i1], 1.0f);
        atomicAdd(&sCnt[i2], 1.0f);
        atomicAdd(&sEnt, h);
    }
    __syncthreads();

    // per-expert column sums of probs (conflict-free: padded stride), then
    // one global float atomic per expert per block
    if (tid < NUM_EXPERTS) {
        float cs = 0.0f;
        for (int r = 0; r < TOK_PER_WG; ++r) cs += sL[r][tid];
        long base = 4l * nTok;
        atomicAdd(&out[base + 64 + tid], cs);        // avg_probs accumulator
        atomicAdd(&out[base + tid], sCnt[tid]);      // expert_counts
        if (tid == 0) atomicAdd(&out[base + 128], sEnt);  // entropy accumulator
    }
}

// ---------------------------------------------------------------------------
// Finalize: means + 64-element sort + gini (single block)
// ---------------------------------------------------------------------------
__global__ __launch_bounds__(64) void router_fini(float* __restrict__ out, int nTok) {
    int e = threadIdx.x;
    long base = 4l * nTok;
    float invN = 1.0f / (float)nTok;
    out[base + 64 + e] *= invN;                 // avg_probs = sum / N
    if (e == 0) {
        out[base + 128] *= invN;                // entropy mean

        float c[NUM_EXPERTS];
        for (int i = 0; i < NUM_EXPERTS; ++i) c[i] = out[base + i];
        // insertion sort ascending
        for (int i = 1; i < NUM_EXPERTS; ++i) {
            float key = c[i];
            int j = i - 1;
            while (j >= 0 && c[j] > key) { c[j + 1] = c[j]; --j; }
            c[j + 1] = key;
        }
        float tot = 0.0f, gsum = 0.0f;
        for (int i = 0; i < NUM_EXPERTS; ++i) tot += c[i];
        for (int i = 1; i <= NUM_EXPERTS; ++i)
            gsum += (2.0f * (float)i - (float)NUM_EXPERTS - 1.0f) * c[i - 1];
        out[base + 129] = gsum / ((float)NUM_EXPERTS * tot + 1e-10f);
    }
}

extern "C" void kernel_launch(void* const* d_in, const int* in_sizes, int n_in,
                              void* d_out, int out_size, void* d_ws, size_t ws_size,
                              hipStream_t stream) {
    const float* x = (const float*)d_in[0];
    const float* W = (const float*)d_in[1];
    float* out = (float*)d_out;
    int nTok = in_sizes[0] / HIDDEN_DIM;   // 16384

    router_init<<<1, 256, 0, stream>>>(out, nTok);
    router_main<<<nTok / TOK_PER_WG, THREADS, 0, stream>>>(x, W, out, nTok);
    router_fini<<<1, 64, 0, stream>>>(out, nTok);
}